// FSPooling_65171833749627
// MI455X (gfx1250) — compile-verified
//
#include <hip/hip_runtime.h>
#include <stdint.h>

// FSPool (fixed degree 32, C=128, 5 pieces):
//   out[i,c] = sum_d sort_desc_d( x[col[i*32+d], c] ) * pwl_weight(c, d)
//
// CDNA5 path: Tensor Data Mover *gather mode* pulls the 32 neighbor rows
// (two 16-row gather descriptors, 16-bit indices) straight into LDS;
// fallback is per-lane global_load_async_to_lds_b128. Per-channel
// in-register bitonic sort (pure VALU), constant-folded PWL weights.

#define CCH 128           // channels
#define DEG 32            // fixed neighborhood size
#define PIECES 5

#if __has_builtin(__builtin_amdgcn_tensor_load_to_lds) && \
    __has_builtin(__builtin_amdgcn_s_wait_tensorcnt)
#define USE_TDM 1
#else
#define USE_TDM 0
#endif

#if USE_TDM
typedef __attribute__((ext_vector_type(4))) unsigned int u32x4;
typedef __attribute__((ext_vector_type(8))) int          i32x8;
typedef __attribute__((ext_vector_type(4))) int          i32x4;

// Gather 16 rows of 128 f32 (8KB) from x into LDS via one TDM descriptor.
// D# bitfields per CDNA5 ISA ch. 8 (gather mode, 16-bit indices).
__device__ __forceinline__ void tdm_gather16(uint64_t gbase, uint32_t lds_addr,
                                             const int* ci, int nrows) {
    u32x4 g0 = {
        0x80000001u,                                   // count=1, gather_mode=1, 16b idx
        lds_addr,                                      // LDS dest (bytes)
        (unsigned)gbase,                               // global_addr[31:0]
        ((unsigned)(gbase >> 32) & 0x01FFFFFFu) | (2u << 30)  // addr[56:32] | type=2
    };
    i32x8 g1 = {
        0x00020000,                                    // wg_mask=0, data_size=4B
        (int)(128u << 16),                             // tensor_dim0=128 (low 16)
        (int)(((unsigned)nrows & 0xFFFFu) << 16),      // tensor_dim1=N (low 16)
        (int)((128u << 16) | (((unsigned)nrows >> 16) & 0xFFFFu)), // tile_dim0=128 | dim1 hi
        16,                                            // tile_dim1 = #valid indices
        128,                                           // tensor_dim0_stride = 128 elems
        0, 0                                           // dim1_stride ignored in gather
    };
    i32x4 g2 = {
        (ci[0]  & 0xFFFF) | (ci[1]  << 16), (ci[2]  & 0xFFFF) | (ci[3]  << 16),
        (ci[4]  & 0xFFFF) | (ci[5]  << 16), (ci[6]  & 0xFFFF) | (ci[7]  << 16)
    };
    i32x4 g3 = {
        (ci[8]  & 0xFFFF) | (ci[9]  << 16), (ci[10] & 0xFFFF) | (ci[11] << 16),
        (ci[12] & 0xFFFF) | (ci[13] << 16), (ci[14] & 0xFFFF) | (ci[15] << 16)
    };
    i32x8 g4 = { 0, 0, 0, 0, 0, 0, 0, 0 };             // unused VADDR4 slot (zero-filled)
    __builtin_amdgcn_tensor_load_to_lds(g0, g1, g2, g3, g4, 0);
}
#endif

__global__ __launch_bounds__(CCH) void fspool_kernel(
    const float* __restrict__ x,
    const int*   __restrict__ col,
    const float* __restrict__ weight,
    float*       __restrict__ out,
    int          n_nodes)
{
    // 32 rows x 128 ch x 4B = 16 KB tile
    __shared__ __align__(16) float smem[DEG * CCH];

    const int node = blockIdx.x;
    const int t    = threadIdx.x;          // channel index 0..127

    const uint32_t lds0  = (uint32_t)(uintptr_t)smem;
    const uint64_t xbase = (uint64_t)(uintptr_t)x;

#if USE_TDM
    // Only wave 0 issues the DMA (TDM ignores EXEC, so make the branch
    // wave-uniform to really skip it on waves 1..3).
    if (__builtin_amdgcn_readfirstlane(t >> 5) == 0) {
        int ci[DEG];
        #pragma unroll
        for (int d = 0; d < DEG; ++d)
            ci[d] = col[(size_t)node * DEG + d];      // block-uniform addresses
        tdm_gather16(xbase, lds0,                ci,      n_nodes);  // rows 0..15
        tdm_gather16(xbase, lds0 + 16 * CCH * 4, ci + 16, n_nodes);  // rows 16..31
        __builtin_amdgcn_s_wait_tensorcnt(0);
    }
#else
    __shared__ int s_col[DEG];
    s_col[t & (DEG - 1)] = col[(size_t)node * DEG + (t & (DEG - 1))]; // redundant, race-free
    __syncthreads();

    // 1024 chunks of 16B; 128 threads x 8 iterations.
    #pragma unroll
    for (int j = 0; j < (DEG * CCH * 4) / (16 * CCH); ++j) {   // 8 iters
        const int k = j * CCH + t;
        const int d = k >> 5;
        const uint32_t goff  = (uint32_t)s_col[d] * (CCH * 4u) + (uint32_t)(k & 31) * 16u;
        const uint32_t laddr = lds0 + (uint32_t)k * 16u;
        asm volatile("global_load_async_to_lds_b128 %0, %1, %2"
                     :: "v"(laddr), "v"(goff), "s"(xbase)
                     : "memory");
    }
#if __has_builtin(__builtin_amdgcn_s_wait_asynccnt)
    __builtin_amdgcn_s_wait_asynccnt(0);
#else
    asm volatile("s_wait_asynccnt 0" ::: "memory");
#endif
#endif

    // Overlap-friendly: fetch this channel's 6 weight knots.
    float w6[PIECES + 1];
    #pragma unroll
    for (int j = 0; j <= PIECES; ++j)
        w6[j] = weight[t * (PIECES + 1) + j];

    __syncthreads();

    // Read our channel column: addr = d*512 + c*4 -> conflict-free banks.
    float v[DEG];
    #pragma unroll
    for (int d = 0; d < DEG; ++d)
        v[d] = smem[d * CCH + t];

    // In-register bitonic sort, descending (240 compare-exchanges).
    #pragma unroll
    for (int k = 2; k <= DEG; k <<= 1) {
        #pragma unroll
        for (int j = k >> 1; j > 0; j >>= 1) {
            #pragma unroll
            for (int i = 0; i < DEG; ++i) {
                const int l = i ^ j;
                if (l > i) {
                    const bool desc = ((i & k) == 0);
                    const float a = v[i], b = v[l];
                    const float mx = fmaxf(a, b), mn = fminf(a, b);
                    v[i] = desc ? mx : mn;
                    v[l] = desc ? mn : mx;
                }
            }
        }
    }

    // Weighted sum; sizes = d/31 is node-independent -> constants fold.
    float acc = 0.0f;
    #pragma unroll
    for (int d = 0; d < DEG; ++d) {
        const float s    = fminf((float)d / (float)(DEG - 1), 1.0f);
        const float fidx = (float)PIECES * s;
        const int   i0   = (int)fidx;
        const float fr   = fidx - (float)i0;
        const int   i1   = (i0 + 1 > PIECES) ? PIECES : i0 + 1;
        acc += v[d] * ((1.0f - fr) * w6[i0] + fr * w6[i1]);
    }

    out[(size_t)node * CCH + t] = acc;
}

extern "C" void kernel_launch(void* const* d_in, const int* in_sizes, int n_in,
                              void* d_out, int out_size, void* d_ws, size_t ws_size,
                              hipStream_t stream) {
    const float* x      = (const float*)d_in[0];
    const int*   edge   = (const int*)d_in[1];   // [2, E] flattened
    const float* weight = (const float*)d_in[2]; // [C, PIECES+1]
    float*       out    = (float*)d_out;

    const int E = in_sizes[1] / 2;
    const int N = in_sizes[0] / CCH;
    const int* col = edge + E;                   // second row of edge_index

    fspool_kernel<<<N, CCH, 0, stream>>>(x, col, weight, out, N);
}